// GeodesicLoss_70746701300159
// MI455X (gfx1250) — compile-verified
//
#include <hip/hip_runtime.h>
#include <math.h>

// float ext-vector types matching CDNA5 WMMA f32 16x16x4 operands:
// A = 16x4 f32 -> 2 VGPRs/lane, B = 4x16 f32 -> 2 VGPRs/lane, C/D = 16x16 f32 -> 8 VGPRs/lane
typedef float v2f __attribute__((ext_vector_type(2)));
typedef float v8f __attribute__((ext_vector_type(8)));
// Tensor DMA descriptor groups (clang-23 6-arg builtin form)
typedef unsigned int u32x4 __attribute__((ext_vector_type(4)));
typedef int i32x4 __attribute__((ext_vector_type(4)));
typedef int i32x8 __attribute__((ext_vector_type(8)));

#define LDSTRIDE 17            // 17 coprime with 64 banks -> conflict-free column access
// A, V, S, T, Y matrices (16 x LDSTRIDE each) + rsq[16] + reduce[16]
#define WAVE_FLOATS (5*16*LDSTRIDE + 16 + 16)
#define WB() __builtin_amdgcn_wave_barrier()

// Issue one TDM tensor load: 16x16 f32 row-major tile at `gaddr` -> LDS byte
// address `lds_byte_addr`, padded +1 DWORD after every 16 DWORDs so the tile
// lands with row stride LDSTRIDE(=17) floats. D# layout per CDNA5 ISA §8.
__device__ inline void tdm_load_tile16(unsigned long long gaddr, unsigned int lds_byte_addr) {
  u32x4 g0;
  g0[0] = 1u;                                        // count=1 (valid descriptor)
  g0[1] = lds_byte_addr;                             // lds_addr [63:32]
  g0[2] = (unsigned int)gaddr;                       // global_addr [95:64]
  g0[3] = (unsigned int)((gaddr >> 32) & 0x01FFFFFFu) | 0x80000000u; // addr[56:32] | type=2
  i32x8 g1;
  g1[0] = (int)((2u << 16) | (1u << 20) | (3u << 22)); // data_size=4B, pad_en, ival=16dw, amt=1dw
  g1[1] = (int)(16u << 16);                          // tensor_dim0 = 16 (bits 79:48)
  g1[2] = (int)(16u << 16);                          // tensor_dim1 = 16 (bits 111:80)
  g1[3] = (int)(16u << 16);                          // tile_dim0 = 16 (bits 127:112)
  g1[4] = (int)16u;                                  // tile_dim1 = 16, tile_dim2 = 0
  g1[5] = (int)16u;                                  // tensor_dim0_stride = 16 (48b, low32)
  g1[6] = (int)(256u << 16);                         // dim0_stride hi=0 | tensor_dim1_stride lo16
  g1[7] = 0;                                         // tensor_dim1_stride hi
  i32x4 gz4 = {0, 0, 0, 0};                          // groups 2/3 unused (2-D tile)
  i32x8 gz8 = {0, 0, 0, 0, 0, 0, 0, 0};              // extra group slot (clang-23 form), unused
  __builtin_amdgcn_tensor_load_to_lds(g0, g1, gz4, gz4, gz8, 0);
}

// Cyclic Jacobi on a 16x16 symmetric matrix held in LDS (row stride LDSTRIDE).
// Wave-cooperative: lanes 0..15 own the "p" slot, lanes 16..31 the "q" slot of
// each rotation, so all 32 lanes are busy. Rotation params are wave-uniform
// (broadcast LDS reads), so branches stay uniform.
template <bool WITHV>
__device__ inline void jacobi16(float* __restrict__ A, float* __restrict__ V, int lane) {
  const int j  = lane & 15;
  const int hi = lane >> 4;
  for (int sw = 0; sw < 6; ++sw) {
    for (int p = 0; p < 15; ++p) {
      for (int q = p + 1; q < 16; ++q) {
        float app = A[p * LDSTRIDE + p];
        float aqq = A[q * LDSTRIDE + q];
        float apq = A[p * LDSTRIDE + q];
        float c = 1.0f, s = 0.0f;
        if (fabsf(apq) > 1e-20f) {              // wave-uniform branch
          float theta = 0.5f * (aqq - app) / apq;
          float t = copysignf(1.0f, theta) /
                    (fabsf(theta) + sqrtf(1.0f + theta * theta));
          c = rsqrtf(1.0f + t * t);
          s = t * c;
        }
        // Row update: rows p and q (lane-half picks which row to write).
        float ap = A[p * LDSTRIDE + j];
        float aq = A[q * LDSTRIDE + j];
        WB();
        if (hi == 0) A[p * LDSTRIDE + j] = c * ap - s * aq;
        else         A[q * LDSTRIDE + j] = s * ap + c * aq;
        WB();
        // Column update: columns p and q.
        float bp = A[j * LDSTRIDE + p];
        float bq = A[j * LDSTRIDE + q];
        WB();
        if (hi == 0) A[j * LDSTRIDE + p] = c * bp - s * bq;
        else         A[j * LDSTRIDE + q] = s * bp + c * bq;
        WB();
        if (WITHV) {
          float vp = V[j * LDSTRIDE + p];
          float vq = V[j * LDSTRIDE + q];
          WB();
          if (hi == 0) V[j * LDSTRIDE + p] = c * vp - s * vq;
          else         V[j * LDSTRIDE + q] = s * vp + c * vq;
          WB();
        }
      }
    }
  }
}

__global__ __launch_bounds__(256)
void geodesic_batch_kernel(const float* __restrict__ yhat,
                           const float* __restrict__ y,
                           float* __restrict__ partial, int nb) {
  __shared__ float smem[8 * WAVE_FLOATS];
  const int lane = threadIdx.x & 31;
  const int wave = threadIdx.x >> 5;
  float* Aw   = &smem[wave * WAVE_FLOATS];
  float* Vw   = Aw + 16 * LDSTRIDE;
  float* Sw   = Vw + 16 * LDSTRIDE;
  float* Tw   = Sw + 16 * LDSTRIDE;
  float* Yw   = Tw + 16 * LDSTRIDE;
  float* rsqA = Yw + 16 * LDSTRIDE;
  float* redA = rsqA + 16;

  // LDS byte offsets of Aw / Yw for the TDM descriptors (smem is the only
  // shared object, so it sits at LDS offset 0).
  const unsigned int ldsAoff = (unsigned int)(wave * WAVE_FLOATS) * 4u;
  const unsigned int ldsYoff = ldsAoff + (unsigned int)(4 * 16 * LDSTRIDE) * 4u;

  const int l  = lane & 15;
  const int hi = lane >> 4;
  const int totalWaves = gridDim.x * 8;

  for (int b = blockIdx.x * 8 + wave; b < nb; b += totalWaves) {  // wave-uniform loop
    const float* Ab = yhat + (size_t)b * 256;
    const float* Yb = y    + (size_t)b * 256;

    // TDM: DMA both 16x16 tiles straight into LDS at stride-17 layout.
    tdm_load_tile16((unsigned long long)(uintptr_t)Ab, ldsAoff);
    tdm_load_tile16((unsigned long long)(uintptr_t)Yb, ldsYoff);

    // Init V = I while the DMA flies.
#pragma unroll
    for (int i = 0; i < 8; ++i) {
      int e = lane * 8 + i;
      int r = e >> 4, cc = e & 15;
      Vw[r * LDSTRIDE + cc] = (r == cc) ? 1.0f : 0.0f;
    }
    __builtin_amdgcn_s_wait_tensorcnt(0);   // both tiles resident in LDS
    WB();

    // Eigendecomposition of yhat: Aw diag -> w, Vw -> eigenvectors.
    jacobi16<true>(Aw, Vw, lane);
    if (lane < 16) rsqA[l] = rsqrtf(fmaxf(Aw[l * LDSTRIDE + l], 1e-30f));
    WB();

    // S = (V * diag(w^-1/2)) * V^T   — 16x16x16 via 4x v_wmma_f32_16x16x4_f32
    {
      v8f acc = {};
#pragma unroll
      for (int k0 = 0; k0 < 16; k0 += 4) {
        int k = k0 + 2 * hi;                       // ISA A/B fragment layout
        v2f a, bb;
        a.x  = Vw[l * LDSTRIDE + k]     * rsqA[k];
        a.y  = Vw[l * LDSTRIDE + k + 1] * rsqA[k + 1];
        bb.x = Vw[l * LDSTRIDE + k];               // (V^T)[k][n] = V[n][k], n = l
        bb.y = Vw[l * LDSTRIDE + k + 1];
        acc = __builtin_amdgcn_wmma_f32_16x16x4_f32(false, a, false, bb,
                                                    (short)0, acc, false, false);
      }
#pragma unroll
      for (int r = 0; r < 8; ++r) Sw[(r + hi * 8) * LDSTRIDE + l] = acc[r];  // C layout
    }
    WB();

    // T = S * Y   (Y tile now resident in LDS via TDM)
    {
      v8f acc = {};
#pragma unroll
      for (int k0 = 0; k0 < 16; k0 += 4) {
        int k = k0 + 2 * hi;
        v2f a, bb;
        a.x  = Sw[l * LDSTRIDE + k];
        a.y  = Sw[l * LDSTRIDE + k + 1];
        bb.x = Yw[k * LDSTRIDE + l];
        bb.y = Yw[(k + 1) * LDSTRIDE + l];
        acc = __builtin_amdgcn_wmma_f32_16x16x4_f32(false, a, false, bb,
                                                    (short)0, acc, false, false);
      }
#pragma unroll
      for (int r = 0; r < 8; ++r) Tw[(r + hi * 8) * LDSTRIDE + l] = acc[r];
    }
    WB();

    // M = T * S  -> overwrite Aw for the second eigensolve
    {
      v8f acc = {};
#pragma unroll
      for (int k0 = 0; k0 < 16; k0 += 4) {
        int k = k0 + 2 * hi;
        v2f a, bb;
        a.x  = Tw[l * LDSTRIDE + k];
        a.y  = Tw[l * LDSTRIDE + k + 1];
        bb.x = Sw[k * LDSTRIDE + l];
        bb.y = Sw[(k + 1) * LDSTRIDE + l];
        acc = __builtin_amdgcn_wmma_f32_16x16x4_f32(false, a, false, bb,
                                                    (short)0, acc, false, false);
      }
#pragma unroll
      for (int r = 0; r < 8; ++r) Aw[(r + hi * 8) * LDSTRIDE + l] = acc[r];
    }
    WB();

    // Eigenvalues of M; per-sample loss = sum_i log(lambda_i)^2
    jacobi16<false>(Aw, nullptr, lane);
    if (lane < 16) {
      float ev = fmaxf(Aw[l * LDSTRIDE + l], 1e-30f);
      float lg = logf(ev);
      redA[l] = lg * lg;
    }
    WB();
    if (lane == 0) {
      float ssum = 0.0f;
      for (int i = 0; i < 16; ++i) ssum += redA[i];   // fixed order -> deterministic
      partial[b] = ssum;
    }
    WB();
  }
}

// Deterministic fixed-order reduction: 256 threads, strided serial sums + tree.
__global__ __launch_bounds__(256)
void geodesic_reduce_kernel(const float* __restrict__ partial, int nb,
                            float* __restrict__ out) {
  __shared__ float sm[256];
  int t = threadIdx.x;
  float s = 0.0f;
  for (int i = t; i < nb; i += 256) s += partial[i];
  sm[t] = s;
  __syncthreads();
  for (int stride = 128; stride > 0; stride >>= 1) {
    if (t < stride) sm[t] += sm[t + stride];
    __syncthreads();
  }
  if (t == 0) out[0] = sqrtf(sm[0]);
}

extern "C" void kernel_launch(void* const* d_in, const int* in_sizes, int n_in,
                              void* d_out, int out_size, void* d_ws, size_t ws_size,
                              hipStream_t stream) {
  const float* yhat = (const float*)d_in[0];
  const float* y    = (const float*)d_in[1];
  float* out        = (float*)d_out;
  float* partial    = (float*)d_ws;           // nb floats of scratch

  int nb = in_sizes[0] / 256;                 // number of 16x16 matrices
  int blocks = (nb + 7) / 8;                  // 8 waves per block, 1+ batch per wave
  if (blocks > 2048) blocks = 2048;
  if (blocks < 1) blocks = 1;

  geodesic_batch_kernel<<<blocks, 256, 0, stream>>>(yhat, y, partial, nb);
  geodesic_reduce_kernel<<<1, 256, 0, stream>>>(partial, nb, out);
}